// SASA_41162966565412
// MI455X (gfx1250) — compile-verified
//
#include <hip/hip_runtime.h>
#include <hip/hip_bf16.h>

typedef __attribute__((ext_vector_type(2))) float v2f;
typedef __attribute__((ext_vector_type(8))) float v8f;

// Problem constants (match reference)
constexpr int CB    = 2;            // batch
constexpr int CC    = 192;          // channels (== ODIM)
constexpr int CH    = 56;
constexpr int CW    = 56;
constexpr int CN    = CH * CW;      // 3136 pixels = 196 * 16 (exact)
constexpr int NT    = CN / 16;      // 196 N-tiles
constexpr int MT    = CC / 16;      // 12  M-tiles
constexpr int MB    = MT / 4;       // 3  M-blocks (64 rows each)
constexpr int HEADS = 6;
constexpr int DH    = CC / HEADS;   // 32
constexpr int KS    = 7;
constexpr int PADC  = 3;
constexpr int KK    = KS * KS;      // 49

// ---------------------------------------------------------------------------
// WMMA f32 16x16x4 GEMM, register-blocked 64x16 per wave:
//   O[m0..m0+63, n0..n0+15] = sum_c W[m,c] * X[c,n]
// The B fragment (x columns, strided loads) is loaded ONCE per K-step and
// reused against 4 A fragments (weight rows, contiguous b64 loads).
// Fragment layouts per ISA 7.12.2:
//   A (16x4):  lane L -> row  M = L&15, K = ((L>>4)*2) + {0,1}
//   B (4x16):  lane L -> col  N = L&15, K = ((L>>4)*2) + {0,1}
//   C/D:       vgpr r -> row  M = ((L>>4)*8) + r,  col N = L&15
// ---------------------------------------------------------------------------
__device__ __forceinline__ void gemm_tile_f32_m4(
    const float* __restrict__ w,   // [CC x CC] row-major
    const float* __restrict__ x,   // [CC x CN] row-major (one batch)
    float* __restrict__ o,         // [CC x CN] row-major (one batch)
    int m0, int n0, int lane)
{
    const int ml   = lane & 15;
    const int koff = (lane >> 4) << 1;

    const float* wrow = w + (m0 + ml) * CC + koff;
    const float* xcol = x + (long)koff * CN + (n0 + ml);

    v8f acc0 = {0.f,0.f,0.f,0.f,0.f,0.f,0.f,0.f};
    v8f acc1 = acc0, acc2 = acc0, acc3 = acc0;

    #pragma unroll 4
    for (int ks = 0; ks < CC; ks += 4) {
        // B fragment: loaded once, reused by 4 WMMAs
        v2f bf;
        bf.x = xcol[0];
        bf.y = xcol[CN];
        xcol += 4 * CN;

        // 4 A fragments: contiguous pairs -> global_load_b64 each
        v2f a0, a1, a2, a3;
        a0.x = wrow[ks];            a0.y = wrow[ks + 1];
        a1.x = wrow[ks + 16 * CC];  a1.y = wrow[ks + 16 * CC + 1];
        a2.x = wrow[ks + 32 * CC];  a2.y = wrow[ks + 32 * CC + 1];
        a3.x = wrow[ks + 48 * CC];  a3.y = wrow[ks + 48 * CC + 1];

        acc0 = __builtin_amdgcn_wmma_f32_16x16x4_f32(false, a0, false, bf,
                                                     (short)0, acc0, false, false);
        acc1 = __builtin_amdgcn_wmma_f32_16x16x4_f32(false, a1, false, bf,
                                                     (short)0, acc1, false, false);
        acc2 = __builtin_amdgcn_wmma_f32_16x16x4_f32(false, a2, false, bf,
                                                     (short)0, acc2, false, false);
        acc3 = __builtin_amdgcn_wmma_f32_16x16x4_f32(false, a3, false, bf,
                                                     (short)0, acc3, false, false);
    }

    float* orow = o + n0 + ml;
    const int mbase = m0 + ((lane >> 4) << 3);
    #pragma unroll
    for (int r = 0; r < 8; ++r) {
        orow[(long)(mbase + r)      * CN] = acc0[r];
        orow[(long)(mbase + r + 16) * CN] = acc1[r];
        orow[(long)(mbase + r + 32) * CN] = acc2[r];
        orow[(long)(mbase + r + 48) * CN] = acc3[r];
    }
}

// Fused Q/K/V projection: 3 GEMMs of 192x192 x 192x3136, per batch.
__global__ __launch_bounds__(256)
void sasa_qkv_proj(const float* __restrict__ x,
                   const float* __restrict__ wq,
                   const float* __restrict__ wk,
                   const float* __restrict__ wv,
                   float* __restrict__ qb,
                   float* __restrict__ kb,
                   float* __restrict__ vb)
{
    const int lane = threadIdx.x & 31;
    const int wave = threadIdx.x >> 5;
    const long tile = (long)blockIdx.x * 8 + wave;

    const long perMat = (long)CB * MB * NT;          // 2*3*196 = 1176
    if (tile >= 3 * perMat) return;                  // whole-wave exit, EXEC stays full

    const int which = (int)(tile / perMat);
    const long t  = tile % perMat;
    const int b   = (int)(t / (MB * NT));
    const int mb  = (int)((t % (MB * NT)) / NT);
    const int nt  = (int)(t % NT);

    const float* w = (which == 0) ? wq : (which == 1) ? wk : wv;
    float*       o = (which == 0) ? qb : (which == 1) ? kb : vb;

    const long boff = (long)b * CC * CN;
    gemm_tile_f32_m4(w, x + boff, o + boff, mb * 64, nt * 16, lane);
}

// Output projection: ctx (192 x 3136 per batch) -> out, with w_proj.
__global__ __launch_bounds__(256)
void sasa_out_proj(const float* __restrict__ ctx,
                   const float* __restrict__ wp,
                   float* __restrict__ out)
{
    const int lane = threadIdx.x & 31;
    const int wave = threadIdx.x >> 5;
    const long tile = (long)blockIdx.x * 8 + wave;

    const long perMat = (long)CB * MB * NT;          // 1176
    if (tile >= perMat) return;

    const int b  = (int)(tile / (MB * NT));
    const int mb = (int)((tile % (MB * NT)) / NT);
    const int nt = (int)(tile % NT);

    const long boff = (long)b * CC * CN;
    gemm_tile_f32_m4(wp, ctx + boff, out + boff, mb * 64, nt * 16, lane);
}

// ---------------------------------------------------------------------------
// Windowed attention core (not WMMA-able: per-pixel key/value sets — the
// sliding 7x7 window means every output pixel has its own K/V matrix, so
// this is a batch of 49x32 GEMVs; 0.24 GFLOP total -> VALU).
// One thread per (b, head, pixel). Channel-major layout => consecutive
// threads (consecutive pixels) produce coalesced k/v reads.
// Reference semantics: zero-padded k/v patches => OOB neighbor score is
// just the positional bias (participates in softmax), v contributes zero.
// ---------------------------------------------------------------------------
__global__ __launch_bounds__(256)
void sasa_attn(const float* __restrict__ qb,
               const float* __restrict__ kb,
               const float* __restrict__ vb,
               const float* __restrict__ pos,   // 13 entries, bias = pos[di+dj]
               float* __restrict__ ctx)
{
    const int tid   = blockIdx.x * blockDim.x + threadIdx.x;
    const int total = CB * HEADS * CN;
    if (tid >= total) return;

    const int pix  = tid % CN;
    const int head = (tid / CN) % HEADS;
    const int b    = tid / (CN * HEADS);
    const int i    = pix / CW;
    const int j    = pix % CW;

    const long base = ((long)b * CC + head * DH) * CN;

    // q vector in registers
    float q[DH];
    const float* qp = qb + base + pix;
    #pragma unroll
    for (int d = 0; d < DH; ++d) q[d] = qp[(long)d * CN];

    // scores
    float s[KK];
    float smax = -1e30f;
    #pragma unroll 7
    for (int kkk = 0; kkk < KK; ++kkk) {
        const int di = kkk / KS, dj = kkk % KS;
        const int ni = i + di - PADC, nj = j + dj - PADC;
        float dot = 0.f;
        if (ni >= 0 && ni < CH && nj >= 0 && nj < CW) {
            const float* kp = kb + base + ni * CW + nj;
            #pragma unroll
            for (int d = 0; d < DH; ++d) dot = fmaf(q[d], kp[(long)d * CN], dot);
        }
        const float sc = dot + pos[di + dj];
        s[kkk] = sc;
        smax = fmaxf(smax, sc);
    }

    // softmax over 49 window positions
    float denom = 0.f;
    #pragma unroll 7
    for (int kkk = 0; kkk < KK; ++kkk) {
        const float e = expf(s[kkk] - smax);
        s[kkk] = e;
        denom += e;
    }
    const float inv = 1.f / denom;

    // context = sum_kk p * v(neighbor)
    float o[DH];
    #pragma unroll
    for (int d = 0; d < DH; ++d) o[d] = 0.f;

    #pragma unroll 7
    for (int kkk = 0; kkk < KK; ++kkk) {
        const int di = kkk / KS, dj = kkk % KS;
        const int ni = i + di - PADC, nj = j + dj - PADC;
        if (ni >= 0 && ni < CH && nj >= 0 && nj < CW) {
            const float p = s[kkk] * inv;
            const float* vp = vb + base + ni * CW + nj;
            #pragma unroll
            for (int d = 0; d < DH; ++d) o[d] = fmaf(p, vp[(long)d * CN], o[d]);
        }
    }

    float* op = ctx + base + pix;
    #pragma unroll
    for (int d = 0; d < DH; ++d) op[(long)d * CN] = o[d];
}

// ---------------------------------------------------------------------------
extern "C" void kernel_launch(void* const* d_in, const int* in_sizes, int n_in,
                              void* d_out, int out_size, void* d_ws, size_t ws_size,
                              hipStream_t stream)
{
    const float* x   = (const float*)d_in[0];
    const float* wq  = (const float*)d_in[1];
    const float* wk  = (const float*)d_in[2];
    const float* wv  = (const float*)d_in[3];
    const float* pos = (const float*)d_in[4];
    const float* wp  = (const float*)d_in[5];
    // d_in[6..9]: kernel_size/padding/stride/num_heads — hardcoded constants.

    float* out = (float*)d_out;

    const long S = (long)CB * CC * CN;   // 1,204,224 floats per tensor
    float* ws  = (float*)d_ws;
    float* qb  = ws;
    float* kb  = ws + S;
    float* vb  = ws + 2 * S;
    float* ctx = ws + 3 * S;

    // 1) Q/K/V projections: 3 * CB * 3 * 196 = 3528 wave-tiles, 8 waves/block
    {
        const long tiles = 3L * CB * MB * NT;      // 3528
        const int blocks = (int)((tiles + 7) / 8); // 441
        sasa_qkv_proj<<<blocks, 256, 0, stream>>>(x, wq, wk, wv, qb, kb, vb);
    }

    // 2) Windowed attention: one thread per (b, head, pixel)
    {
        const int total  = CB * HEADS * CN;        // 37632
        const int blocks = (total + 255) / 256;    // 147
        sasa_attn<<<blocks, 256, 0, stream>>>(qb, kb, vb, pos, ctx);
    }

    // 3) Output projection into d_out
    {
        const long tiles = (long)CB * MB * NT;     // 1176
        const int blocks = (int)((tiles + 7) / 8); // 147
        sasa_out_proj<<<blocks, 256, 0, stream>>>(ctx, wp, out);
    }
}